// GptOssExperts_27857157882043
// MI455X (gfx1250) — compile-verified
//
#include <hip/hip_runtime.h>
#include <math.h>

typedef float v2f __attribute__((ext_vector_type(2)));
typedef float v8f __attribute__((ext_vector_type(8)));

#define NE     32      // experts
#define HD     512     // hidden
#define ED     512     // expert dim
#define NTOK   256     // tokens
#define NSLOT  512     // tokens * topk
#define ALPHA  1.702f
#define LIMIT  7.0f
#define XPITCH (HD + 4)   // LDS pitch: 516 -> bank stride 4/row, conflict-free

// ---------------------------------------------------------------------------
// Kernel 1: gate_up GEMM + activation, per (expert, 128-col chunk of 2*ED)
//   fused[slot][f] for f in [chunk*64, chunk*64+64)
// ---------------------------------------------------------------------------
__global__ __launch_bounds__(256) void moe_gemm1(
    const float* __restrict__ hs,   // [256,512]
    const int*   __restrict__ ri,   // [256,2]
    const float* __restrict__ w1,   // [32,512,1024]
    const float* __restrict__ b1,   // [32,1024]
    float*       __restrict__ fused)// ws: [512,512]
{
    __shared__ int   s_list[NSLOT];
    __shared__ int   s_cnt;
    __shared__ int   s_slot[16];
    __shared__ float s_x[16 * XPITCH];

    const int e     = blockIdx.x;
    const int chunk = blockIdx.y;          // 0..7, 128 gate_up cols each
    const int tid   = threadIdx.x;
    const int wave  = tid >> 5;
    const int lane  = tid & 31;
    const int nn    = lane & 15;
    const int half  = lane >> 4;

    if (tid == 0) s_cnt = 0;
    __syncthreads();
    for (int s = tid; s < NSLOT; s += 256) {
        if (ri[s] == e) {
            int p = atomicAdd(&s_cnt, 1);  // order-independent: rows are independent
            s_list[p] = s;
        }
    }
    __syncthreads();
    const int cnt = s_cnt;

    const int    cbase = chunk * 128 + wave * 16;        // this wave's 16 gu cols
    const float* w1e   = w1 + (size_t)e * HD * (2 * ED);
    const float* bptr  = w1e + cbase + nn;
    const float  bias  = b1[e * (2 * ED) + cbase + nn];

    for (int t0 = 0; t0 < cnt; t0 += 16) {
        if (tid < 16) {
            int idx = t0 + tid;
            s_slot[tid] = (idx < cnt) ? s_list[idx] : -1;
        }
        __syncthreads();
        // cooperative load of 16 x 512 X tile (float4)
        for (int i = tid; i < 16 * (HD / 4); i += 256) {
            int r   = i >> 7;              // HD/4 == 128
            int c4  = (i & 127) << 2;
            int sl  = s_slot[r];
            int tok = (sl >= 0) ? (sl >> 1) : 0;
            float4 v = *(const float4*)(hs + (size_t)tok * HD + c4);
            *(float4*)(&s_x[r * XPITCH + c4]) = v;
        }
        __syncthreads();

        v8f acc = {0.f, 0.f, 0.f, 0.f, 0.f, 0.f, 0.f, 0.f};
        #pragma unroll 8
        for (int k0 = 0; k0 < HD; k0 += 4) {
            const int ka = k0 + 2 * half;
            // A frag: X[M=nn][ka], X[M=nn][ka+1]
            v2f a = *(const v2f*)(&s_x[nn * XPITCH + ka]);
            // B frag: W1[ka][cbase+nn], W1[ka+1][cbase+nn]
            v2f b;
            b.x = bptr[(size_t)(ka + 0) * (2 * ED)];
            b.y = bptr[(size_t)(ka + 1) * (2 * ED)];
            acc = __builtin_amdgcn_wmma_f32_16x16x4_f32(
                false, a, false, b, (short)0, acc, false, false);
        }

        // activation: even col = gate, odd col = up (adjacent lanes)
        #pragma unroll
        for (int j = 0; j < 8; ++j) {
            float v  = acc[j] + bias;
            float nb = __shfl_xor(v, 1, 32);   // all lanes participate
            if ((nn & 1) == 0) {
                float g   = fminf(v, LIMIT);
                float u   = fminf(fmaxf(nb, -LIMIT), LIMIT);
                // glu = g * sigmoid(ALPHA*g) via fast exp + rcp (1-2 ulp)
                float den = 1.0f + __expf(-ALPHA * g);
                float glu = g * __builtin_amdgcn_rcpf(den);
                float fz  = (u + 1.0f) * glu;
                int   m   = j + 8 * half;
                int   sl  = s_slot[m];
                if (sl >= 0) {
                    int f = (cbase + nn) >> 1;   // fused feature index
                    fused[(size_t)sl * ED + f] = fz;
                }
            }
        }
        __syncthreads();
    }
}

// ---------------------------------------------------------------------------
// Kernel 2: down GEMM, per (expert, 128-col chunk of HD)
//   scr2[slot][d] = rw[t,e] * (fused[slot,:] . W2[e,:,d] + b2[e,d])
// ---------------------------------------------------------------------------
__global__ __launch_bounds__(256) void moe_gemm2(
    const int*   __restrict__ ri,    // [256,2]
    const float* __restrict__ rw,    // [256,32]
    const float* __restrict__ w2,    // [32,512,512]
    const float* __restrict__ b2,    // [32,512]
    const float* __restrict__ fused, // ws: [512,512]
    float*       __restrict__ scr2)  // ws: [512,512]
{
    __shared__ int   s_list[NSLOT];
    __shared__ int   s_cnt;
    __shared__ int   s_slot[16];
    __shared__ float s_wgt[16];
    __shared__ float s_a[16 * XPITCH];

    const int e     = blockIdx.x;
    const int chunk = blockIdx.y;          // 0..3, 128 out cols each
    const int tid   = threadIdx.x;
    const int wave  = tid >> 5;
    const int lane  = tid & 31;
    const int nn    = lane & 15;
    const int half  = lane >> 4;

    if (tid == 0) s_cnt = 0;
    __syncthreads();
    for (int s = tid; s < NSLOT; s += 256) {
        if (ri[s] == e) {
            int p = atomicAdd(&s_cnt, 1);
            s_list[p] = s;
        }
    }
    __syncthreads();
    const int cnt = s_cnt;

    const int    cbase = chunk * 128 + wave * 16;
    const float* w2e   = w2 + (size_t)e * ED * HD;
    const float* bptr  = w2e + cbase + nn;
    const float  bias  = b2[e * HD + cbase + nn];

    for (int t0 = 0; t0 < cnt; t0 += 16) {
        if (tid < 16) {
            int idx = t0 + tid;
            int sl  = (idx < cnt) ? s_list[idx] : -1;
            s_slot[tid] = sl;
            s_wgt[tid]  = (sl >= 0) ? rw[(size_t)(sl >> 1) * NE + e] : 0.f;
        }
        __syncthreads();
        // cooperative load of 16 x 512 fused tile (float4)
        for (int i = tid; i < 16 * (ED / 4); i += 256) {
            int r  = i >> 7;
            int c4 = (i & 127) << 2;
            int sl = s_slot[r];
            int sr = (sl >= 0) ? sl : 0;
            float4 v = *(const float4*)(fused + (size_t)sr * ED + c4);
            *(float4*)(&s_a[r * XPITCH + c4]) = v;
        }
        __syncthreads();

        v8f acc = {0.f, 0.f, 0.f, 0.f, 0.f, 0.f, 0.f, 0.f};
        #pragma unroll 8
        for (int k0 = 0; k0 < ED; k0 += 4) {
            const int ka = k0 + 2 * half;
            v2f a = *(const v2f*)(&s_a[nn * XPITCH + ka]);
            v2f b;
            b.x = bptr[(size_t)(ka + 0) * HD];
            b.y = bptr[(size_t)(ka + 1) * HD];
            acc = __builtin_amdgcn_wmma_f32_16x16x4_f32(
                false, a, false, b, (short)0, acc, false, false);
        }

        #pragma unroll
        for (int j = 0; j < 8; ++j) {
            int m  = j + 8 * half;
            int sl = s_slot[m];
            if (sl >= 0) {
                scr2[(size_t)sl * HD + cbase + nn] = s_wgt[m] * (acc[j] + bias);
            }
        }
        __syncthreads();
    }
}

// ---------------------------------------------------------------------------
// Kernel 3: reduce the two topk slots per token
// ---------------------------------------------------------------------------
__global__ __launch_bounds__(256) void moe_reduce(
    const float* __restrict__ scr2, float* __restrict__ out)
{
    int i = blockIdx.x * blockDim.x + threadIdx.x;
    if (i < NTOK * HD) {
        int t = i / HD;
        int d = i - t * HD;
        out[i] = scr2[(size_t)(2 * t) * HD + d] + scr2[(size_t)(2 * t + 1) * HD + d];
    }
}

extern "C" void kernel_launch(void* const* d_in, const int* in_sizes, int n_in,
                              void* d_out, int out_size, void* d_ws, size_t ws_size,
                              hipStream_t stream) {
    const float* hs = (const float*)d_in[0];
    const int*   ri = (const int*)  d_in[1];
    const float* rw = (const float*)d_in[2];
    const float* w1 = (const float*)d_in[3];
    const float* b1 = (const float*)d_in[4];
    const float* w2 = (const float*)d_in[5];
    const float* b2 = (const float*)d_in[6];
    float* out   = (float*)d_out;
    float* fused = (float*)d_ws;                 // [512,512] fp32 = 1 MB
    float* scr2  = fused + (size_t)NSLOT * ED;   // [512,512] fp32 = 1 MB

    moe_gemm1<<<dim3(NE, 8), 256, 0, stream>>>(hs, ri, w1, b1, fused);
    moe_gemm2<<<dim3(NE, 4), 256, 0, stream>>>(ri, rw, w2, b2, fused, scr2);
    moe_reduce<<<(NTOK * HD + 255) / 256, 256, 0, stream>>>(scr2, out);
}